// MinibatchDiscrimination_37194416783543
// MI455X (gfx1250) — compile-verified
//
#include <hip/hip_runtime.h>

// ---------------------------------------------------------------------------
// MinibatchDiscrimination for MI455X (gfx1250, wave32):
//   Phase 1: Ms = x @ W (2048^3 fp32) via V_WMMA_F32_16X16X4_F32.
//            Double-buffered LDS tiles filled with GLOBAL_LOAD_ASYNC_TO_LDS_B128
//            (ASYNCcnt-tracked DMA) so tile k+1 streams in under tile k's WMMAs.
//   Phase 2: per-batch 128x128 pairwise L1 + exp(-d) row-sum, LDS broadcast.
// ---------------------------------------------------------------------------

typedef float v2f __attribute__((ext_vector_type(2)));
typedef float v8f __attribute__((ext_vector_type(8)));

#define BATCH   2048
#define FDIM    2048
#define NDIM    2048   // KERNELS * DIMS
#define NKERN   128
#define KDIMS   16
#define TM      128
#define TN      128
#define TK      32
#define LDPA    36     // A tile LDS row stride (floats): 16B-aligned rows, conflict-free frags
#define LDPB    136    // B tile LDS row stride (floats): +2 rows => +16 banks (half-wave split)
#define NTILES  (FDIM / TK)

// Async global->LDS DMA, GVS mode: SGPR64 base + VGPR32 byte offset.
// vdst = per-lane LDS byte address, tracked by ASYNCcnt.
__device__ __forceinline__ void async_copy_b128(uint32_t lds_byte,
                                                uint32_t goff_byte,
                                                const void* sbase)
{
    asm volatile("global_load_async_to_lds_b128 %0, %1, %2"
                 :: "v"(lds_byte), "v"(goff_byte), "s"(sbase)
                 : "memory");
}

// Issue one K-tile (A: 128x32, B: 32x128) = 8 async b128 ops per thread.
__device__ __forceinline__ void stage_tile(const float* __restrict__ x,
                                           const float* __restrict__ W,
                                           int tid, int mBase, int nBase, int k0,
                                           float* ldsA, float* ldsB)
{
    #pragma unroll
    for (int it = 0; it < 4; ++it) {                 // A: 1024 float4 slots
        const int idx = tid + it * 256;
        const int row = idx >> 3;                    // 8 float4 per row
        const int kq  = (idx & 7) << 2;
        const uint32_t go = (uint32_t)(((mBase + row) * FDIM + k0 + kq) * 4);
        const uint32_t lo = (uint32_t)(size_t)(ldsA + row * LDPA + kq);
        async_copy_b128(lo, go, (const void*)x);
    }
    #pragma unroll
    for (int it = 0; it < 4; ++it) {                 // B: 1024 float4 slots, row-major
        const int idx = tid + it * 256;
        const int kr  = idx >> 5;                    // 0..31
        const int c4  = (idx & 31) << 2;             // 0..124
        const uint32_t go = (uint32_t)(((k0 + kr) * NDIM + nBase + c4) * 4);
        const uint32_t lo = (uint32_t)(size_t)(ldsB + kr * LDPB + c4);
        async_copy_b128(lo, go, (const void*)W);
    }
}

__global__ __launch_bounds__(256)
void mbd_gemm_wmma_f32(const float* __restrict__ x,
                       const float* __restrict__ W,
                       float* __restrict__ Ms)
{
    __shared__ float ldsA[2][TM * LDPA];   // 2 x 18KB
    __shared__ float ldsB[2][TK * LDPB];   // 2 x 17KB

    const int tid   = threadIdx.x;
    const int lane  = tid & 31;
    const int llo   = lane & 15;           // row/col within 16x16 fragment
    const int khalf = lane >> 4;           // 0: K={0,1}, 1: K={2,3}
    const int w     = tid >> 5;            // wave id 0..7
    const int wm    = w & 3;               // 4 wave-rows  (32 rows each)
    const int wn    = w >> 2;              // 2 wave-cols  (64 cols each)
    const int mBase = blockIdx.y * TM;
    const int nBase = blockIdx.x * TN;

    v8f acc[2][4] = {};                    // 2 (M) x 4 (N) 16x16 f32 accumulators

    // Prologue: stage tile 0 into buffer 0.
    stage_tile(x, W, tid, mBase, nBase, 0, ldsA[0], ldsB[0]);

    for (int kt = 0; kt < NTILES; ++kt) {
        const int cur = kt & 1;
        if (kt + 1 < NTILES) {
            // Stream next tile into the other buffer, then wait until only
            // those 8 ops remain outstanding => current tile's 8 ops landed.
            stage_tile(x, W, tid, mBase, nBase, (kt + 1) * TK,
                       ldsA[cur ^ 1], ldsB[cur ^ 1]);
            asm volatile("s_wait_asynccnt 0x8" ::: "memory");
        } else {
            asm volatile("s_wait_asynccnt 0x0" ::: "memory");
        }
        __syncthreads();                   // all waves' async data visible

        const float* A = ldsA[cur];
        const float* B = ldsB[cur];
        #pragma unroll
        for (int kk = 0; kk < TK; kk += 4) {
            const int ka = kk + 2 * khalf;
            v2f af[2], bf[4];
            #pragma unroll
            for (int mi = 0; mi < 2; ++mi)
                af[mi] = *(const v2f*)(A + (wm * 32 + mi * 16 + llo) * LDPA + ka);
            #pragma unroll
            for (int ni = 0; ni < 4; ++ni) {
                const int col = wn * 64 + ni * 16 + llo;
                v2f t;
                t.x = B[(ka + 0) * LDPB + col];   // fuses to ds_load_2addr_b32
                t.y = B[(ka + 1) * LDPB + col];
                bf[ni] = t;
            }
            #pragma unroll
            for (int mi = 0; mi < 2; ++mi)
                #pragma unroll
                for (int ni = 0; ni < 4; ++ni)
                    acc[mi][ni] = __builtin_amdgcn_wmma_f32_16x16x4_f32(
                        false, af[mi], false, bf[ni],
                        (short)0, acc[mi][ni], false, false);
        }
        __syncthreads();                   // done reading before buffer reuse
    }

    // Write back: VGPR r holds (M = r + 8*khalf, N = llo).
    #pragma unroll
    for (int mi = 0; mi < 2; ++mi) {
        #pragma unroll
        for (int ni = 0; ni < 4; ++ni) {
            const int col = nBase + wn * 64 + ni * 16 + llo;
            #pragma unroll
            for (int r = 0; r < 8; ++r) {
                const int row = mBase + wm * 32 + mi * 16 + r + 8 * khalf;
                Ms[(size_t)row * NDIM + col] = acc[mi][ni][r];
            }
        }
    }
}

// ---------------------------------------------------------------------------
// Phase 2: one block per batch b. Thread i owns kernel row i; inner loop reads
// lds[j*16+d] (same address wave-wide -> LDS broadcast, conflict-free).
// ---------------------------------------------------------------------------
__global__ __launch_bounds__(128)
void mbd_pairwise(const float* __restrict__ Ms, float* __restrict__ out)
{
    __shared__ float sm[NKERN * KDIMS];    // 8 KB

    const int b = blockIdx.x;
    const int i = threadIdx.x;             // 0..127
    const float* Mb = Ms + (size_t)b * (NKERN * KDIMS);

    #pragma unroll
    for (int t = 0; t < 4; ++t) {
        const int idx = (i + t * 128) << 2;
        *(float4*)(sm + idx) = *(const float4*)(Mb + idx);
    }
    __syncthreads();

    float mi[KDIMS];
    #pragma unroll
    for (int d = 0; d < KDIMS; ++d) mi[d] = sm[i * KDIMS + d];

    float accv = 0.0f;
    #pragma unroll 4
    for (int j = 0; j < NKERN; ++j) {
        float norm = 0.0f;
        #pragma unroll
        for (int d = 0; d < KDIMS; ++d)
            norm += fabsf(mi[d] - sm[j * KDIMS + d]);
        accv += __expf(-norm);
    }
    out[(size_t)b * NKERN + i] = accv;
}

extern "C" void kernel_launch(void* const* d_in, const int* in_sizes, int n_in,
                              void* d_out, int out_size, void* d_ws, size_t ws_size,
                              hipStream_t stream)
{
    (void)in_sizes; (void)n_in; (void)out_size; (void)ws_size;
    const float* x = (const float*)d_in[0];
    const float* W = (const float*)d_in[1];
    float* out = (float*)d_out;
    float* Ms  = (float*)d_ws;             // 2048*2048*4 = 16 MB scratch

    dim3 gridG(NDIM / TN, BATCH / TM);     // 16 x 16 workgroups
    mbd_gemm_wmma_f32<<<gridG, 256, 0, stream>>>(x, W, Ms);
    mbd_pairwise<<<BATCH, NKERN, 0, stream>>>(Ms, out);
}